// CAREPositionEncoding_30494267802220
// MI455X (gfx1250) — compile-verified
//
#include <hip/hip_runtime.h>

typedef float v2f __attribute__((ext_vector_type(2)));
typedef float v8f __attribute__((ext_vector_type(8)));

#define MV 16
#define MAXLEN 16384

__device__ constexpr int popc4(int v) { return (v & 1) + ((v >> 1) & 1) + ((v >> 2) & 1) + ((v >> 3) & 1); }

// sign of e_a * e_b in Cl(4,0): C[a,b,a^b] = gpsign(a,b)
__device__ constexpr float gpsign(int a, int b) {
    int s = 0;
    for (int t = a >> 1; t; t >>= 1) s += popc4(t & b);
    return (s & 1) ? -1.0f : 1.0f;
}
__device__ constexpr float bivsign(int k) { return (popc4(k) == 2) ? -1.0f : 1.0f; }

// Transposed GP for a 16-position tile:
//   Out^T(16x16) = Cmat^T(16x256) @ Z^T(256x16),  Z[p, i*16+j] = Af[p,i]*Bf[p,j]
// A operand = constant Cayley chunks from the block-shared LDS table,
// B operand = Z chunks built from per-lane registers.
// Lane layout: kl = lane&15 (M=component for A / N=position for B), hi = lane>=16.
// f32 A/B operand layout: element v on lane-half h corresponds to K = v + 2h,
// chunk c covers rows m = 4c..4c+3 -> i = c>>2, j = 4*(c&3) + v + 2h.
__device__ inline v8f gp_via_wmma_t(const float* __restrict__ ctab,
                                    const float (&Af)[16], const float (&Bf)[16],
                                    int lane, bool hi, v8f acc) {
#pragma unroll
    for (int c = 0; c < 64; ++c) {
        const int i  = c >> 2;
        const int j0 = (c & 3) << 2;
        const v2f aC = *(const v2f*)(ctab + c * 64 + lane * 2);   // Cayley^T chunk
        v2f zB;                                                   // Z^T chunk
        zB[0] = Af[i] * (hi ? Bf[j0 + 2] : Bf[j0 + 0]);
        zB[1] = Af[i] * (hi ? Bf[j0 + 3] : Bf[j0 + 1]);
        acc = __builtin_amdgcn_wmma_f32_16x16x4_f32(false, aC, false, zB,
                                                    (short)0, acc, false, false);
    }
    return acc;
}

__global__ __launch_bounds__(256)
void care_rotor_kernel(const float* __restrict__ x,
                       const int*   __restrict__ pos,
                       const float* __restrict__ bx,
                       const float* __restrict__ by,
                       const float* __restrict__ bz,
                       const float* __restrict__ bw,
                       const float* __restrict__ theta,
                       float* __restrict__ out,
                       int ntiles) {
    // Block-shared Cayley^T operand table: (chunk c, lane) -> 2 floats. 16 KB.
    __shared__ float ctab[64 * 32 * 2];

    const int lane = threadIdx.x & 31;
    const int wave = threadIdx.x >> 5;
    const int kl   = lane & 15;
    const bool hi  = lane >= 16;

    // Wave 0 builds the table once per block; everyone else waits at the barrier.
    if (wave == 0) {
#pragma unroll
        for (int c = 0; c < 64; ++c) {
            const int i  = c >> 2;
            const int j0 = (c & 3) << 2;
            const float e0lo = (kl == (i ^ (j0 + 0))) ? gpsign(i, j0 + 0) : 0.0f;
            const float e1lo = (kl == (i ^ (j0 + 1))) ? gpsign(i, j0 + 1) : 0.0f;
            const float e0hi = (kl == (i ^ (j0 + 2))) ? gpsign(i, j0 + 2) : 0.0f;
            const float e1hi = (kl == (i ^ (j0 + 3))) ? gpsign(i, j0 + 3) : 0.0f;
            v2f e;
            e[0] = hi ? e0hi : e0lo;
            e[1] = hi ? e1hi : e1lo;
            *(v2f*)(ctab + c * 64 + lane * 2) = e;
        }
    }
    __syncthreads();

    const int tile = blockIdx.x * 8 + wave;
    if (tile >= ntiles) return;               // whole-wave exit; EXEC stays all-ones

    const long pbase = (long)tile * 16;
    const long p     = pbase + kl;            // this lane's position (both halves)

    // ---- load x[p, 0..15] ----
    float xv[16];
    {
        const float4* xp = (const float4*)(x + p * MV);
        float4 q[4];
#pragma unroll
        for (int t = 0; t < 4; ++t) q[t] = xp[t];
#pragma unroll
        for (int t = 0; t < 4; ++t) {
            xv[4 * t + 0] = q[t].x; xv[4 * t + 1] = q[t].y;
            xv[4 * t + 2] = q[t].z; xv[4 * t + 3] = q[t].w;
        }
    }

    // ---- gather theta and plane coefficients ----
    int pc = pos[p];
    pc = pc < 0 ? 0 : (pc > MAXLEN - 1 ? MAXLEN - 1 : pc);
    const float4 th = *(const float4*)(theta + (long)pc * 4);
    float phi[4];
    phi[0] = 0.5f * th.x * bx[3];
    phi[1] = 0.5f * th.y * by[5];
    phi[2] = 0.5f * th.z * bz[9];
    phi[3] = 0.5f * th.w * bw[6];

    // ---- rotor = prod_plane (cos(phi) e0 + sin(phi) e_blade), blades (3,5,9,6) ----
    float r[16];
#pragma unroll
    for (int k = 0; k < 16; ++k) r[k] = (k == 0) ? 1.0f : 0.0f;
    const int blades[4] = {3, 5, 9, 6};
#pragma unroll
    for (int pl = 0; pl < 4; ++pl) {
        float sn, cs;
        __sincosf(phi[pl], &sn, &cs);
        const int b = blades[pl];
        float nr[16];
#pragma unroll
        for (int k = 0; k < 16; ++k)
            nr[k] = cs * r[k] + sn * gpsign(k ^ b, b) * r[k ^ b];
#pragma unroll
        for (int k = 0; k < 16; ++k) r[k] = nr[k];
    }

    // reverse: negate bivector components
    float rrev[16];
#pragma unroll
    for (int k = 0; k < 16; ++k) rrev[k] = r[k] * bivsign(k);

    // ---- GP1 (transposed): Rx^T = Cmat^T @ outer(r, x)^T ----
    v8f acc;
#pragma unroll
    for (int v = 0; v < 8; ++v) acc[v] = 0.0f;
    acc = gp_via_wmma_t(ctab, r, xv, lane, hi, acc);

    // D layout: lane holds components (v + 8*hi) of position kl; fetch the other
    // 8 components from lane^16 — no LDS staging, no barrier.
    float rxv[16];
#pragma unroll
    for (int v = 0; v < 8; ++v) {
        const float other = __shfl_xor((float)acc[v], 16, 32);
        rxv[v]     = hi ? other  : acc[v];
        rxv[v + 8] = hi ? acc[v] : other;
    }

    // ---- GP2 (transposed): out^T = Cmat^T @ outer(Rx, Rrev)^T ----
    v8f acc2;
#pragma unroll
    for (int v = 0; v < 8; ++v) acc2[v] = 0.0f;
    acc2 = gp_via_wmma_t(ctab, rxv, rrev, lane, hi, acc2);

    // ---- store: lane holds 8 consecutive components of row (pbase+kl) ----
    {
        float* po = out + (pbase + kl) * MV + (hi ? 8 : 0);
        float4 q0, q1;
        q0.x = acc2[0]; q0.y = acc2[1]; q0.z = acc2[2]; q0.w = acc2[3];
        q1.x = acc2[4]; q1.y = acc2[5]; q1.z = acc2[6]; q1.w = acc2[7];
        *(float4*)(po + 0) = q0;
        *(float4*)(po + 4) = q1;
    }
}

extern "C" void kernel_launch(void* const* d_in, const int* in_sizes, int n_in,
                              void* d_out, int out_size, void* d_ws, size_t ws_size,
                              hipStream_t stream) {
    const float* x     = (const float*)d_in[0];
    const int*   pos   = (const int*)d_in[1];
    const float* bx    = (const float*)d_in[2];
    const float* by    = (const float*)d_in[3];
    const float* bz    = (const float*)d_in[4];
    const float* bw    = (const float*)d_in[5];
    const float* theta = (const float*)d_in[6];
    float* out = (float*)d_out;

    const int npos   = in_sizes[1];          // B * L = 262144
    const int ntiles = npos / 16;            // 16 positions per wave
    const int blocks = (ntiles + 7) / 8;     // 8 waves (256 threads) per block

    hipLaunchKernelGGL(care_rotor_kernel, dim3(blocks), dim3(256), 0, stream,
                       x, pos, bx, by, bz, bw, theta, out, ntiles);
}